// Loss_78271484002528
// MI455X (gfx1250) — compile-verified
//
#include <hip/hip_runtime.h>
#include <math.h>

typedef __attribute__((ext_vector_type(2))) float v2f;
typedef __attribute__((ext_vector_type(8))) float v8f;

constexpr int BATCH          = 4;
constexpr int NPTS           = 8192;  // N == M == 8192
constexpr int ROWS_PER_WAVE  = 32;    // 2 WMMA row-tiles of 16
constexpr int ROWS_PER_BLOCK = 256;   // 8 waves * 32 rows
constexpr int CHUNK          = 512;   // g points staged in LDS per block iter

__global__ void zero_out_kernel(float* out) { out[0] = 0.0f; }

// Min that lowers to bare v_min_num_f32 (no canonicalize) and can dual-issue.
#if __has_builtin(__builtin_fminimum_numf)
__device__ __forceinline__ void vmin(float& acc, float v) {
    acc = __builtin_fminimum_numf(acc, v);
}
#else
__device__ __forceinline__ void vmin(float& acc, float v) {
    asm("v_min_num_f32 %0, %0, %1" : "+v"(acc) : "v"(v));
}
#endif

// dir==0: rows = pcd_pred (d1 = min over gt); dir==1: rows = gt (d2 = min over pred)
__global__ __launch_bounds__(256)
void chamfer_dir_kernel(const float* __restrict__ pcd_pred,
                        const float* __restrict__ gt,
                        float* __restrict__ out)
{
    const int b   = blockIdx.y;
    const int dir = blockIdx.z;
    const float* __restrict__ R = (dir == 0 ? pcd_pred : gt) + (size_t)b * NPTS * 3;
    const float* __restrict__ G = (dir == 0 ? gt : pcd_pred) + (size_t)b * NPTS * 3;

    const int tid  = threadIdx.x;
    const int wave = tid >> 5;
    const int l16  = tid & 15;
    const int hi   = (tid >> 4) & 1;   // which 16-lane half of the wave32

    // Staged B operands: per g point j, {-2gx,-2gy} at [2j], {-2gz,|g|^2} at [2j+1].
    __shared__ v2f   lds_b[CHUNK * 2];   // 8 KB
    __shared__ float lds_red[256];

    const int row_base = blockIdx.x * ROWS_PER_BLOCK + wave * ROWS_PER_WAVE;

    // ---- Build A operands: 16x4 f32, rows = points, K = [x, y, z, 1] ----
    // Layout (ISA 7.12.2): lane l holds row M = l%16; VGPR0 = K0|K2, VGPR1 = K1|K3.
    v2f a[2];
#pragma unroll
    for (int t = 0; t < 2; ++t) {
        const int row = row_base + t * 16 + l16;
        const float px = R[row * 3 + 0];
        const float py = R[row * 3 + 1];
        const float pz = R[row * 3 + 2];
        a[t][0] = hi ? pz   : px;   // K=2 : K=0
        a[t][1] = hi ? 1.0f : py;   // K=3 : K=1
    }

    // Running per-lane min of (|g|^2 - 2 p.g) over all g tiles.
    float mins[2][8];
#pragma unroll
    for (int t = 0; t < 2; ++t)
#pragma unroll
        for (int v = 0; v < 8; ++v) mins[t][v] = 3.0e38f;

    for (int cbase = 0; cbase < NPTS; cbase += CHUNK) {
        __syncthreads();
        // Stage CHUNK g-points into LDS, pre-augmented to (-2gx,-2gy,-2gz,|g|^2).
        for (int i = tid; i < CHUNK; i += 256) {
            const float gx = G[(size_t)(cbase + i) * 3 + 0];
            const float gy = G[(size_t)(cbase + i) * 3 + 1];
            const float gz = G[(size_t)(cbase + i) * 3 + 2];
            const float gn = gx * gx + gy * gy + gz * gz;
            v2f lo, hi2;
            lo[0]  = -2.0f * gx;  lo[1]  = -2.0f * gy;
            hi2[0] = -2.0f * gz;  hi2[1] = gn;
            lds_b[i * 2 + 0] = lo;
            lds_b[i * 2 + 1] = hi2;
        }
        __syncthreads();

#pragma unroll 4
        for (int ct = 0; ct < CHUNK / 16; ++ct) {
            // B operand: lane (l16, hi) -> one ds_load_b64, zero VALU.
            // hi=0: {K0,K1} = {-2gx,-2gy}; hi=1: {K2,K3} = {-2gz,|g|^2}.
            const v2f bb = lds_b[(ct * 16 + l16) * 2 + hi];

#pragma unroll
            for (int t = 0; t < 2; ++t) {
                v8f c = {};
                // D = A x B (C = 0): 16x16 tile of (|g|^2 - 2 p.g)
                c = __builtin_amdgcn_wmma_f32_16x16x4_f32(
                        false, a[t], false, bb, (short)0, c, false, false);
#pragma unroll
                for (int v = 0; v < 8; ++v)
                    vmin(mins[t][v], c[v]);
            }
        }
    }

    // ---- Min across the 16 lanes of each half (columns N of the tile) ----
#pragma unroll
    for (int t = 0; t < 2; ++t) {
#pragma unroll
        for (int v = 0; v < 8; ++v) {
            float x = mins[t][v];
            vmin(x, __shfl_xor(x, 1));
            vmin(x, __shfl_xor(x, 2));
            vmin(x, __shfl_xor(x, 4));
            vmin(x, __shfl_xor(x, 8));
            mins[t][v] = x;
        }
    }

    // C-layout: VGPR v, half hi -> row M = v + 8*hi. Lane l16<8 finishes row (l16 + 8*hi).
    float partial = 0.0f;
    if (l16 < 8) {
#pragma unroll
        for (int t = 0; t < 2; ++t) {
            float s = mins[t][0];
            s = (l16 == 1) ? mins[t][1] : s;
            s = (l16 == 2) ? mins[t][2] : s;
            s = (l16 == 3) ? mins[t][3] : s;
            s = (l16 == 4) ? mins[t][4] : s;
            s = (l16 == 5) ? mins[t][5] : s;
            s = (l16 == 6) ? mins[t][6] : s;
            s = (l16 == 7) ? mins[t][7] : s;
            const int row = row_base + t * 16 + l16 + 8 * hi;
            const float px = R[row * 3 + 0];
            const float py = R[row * 3 + 1];
            const float pz = R[row * 3 + 2];
            const float pn = px * px + py * py + pz * pz;
            const float d  = fmaxf(s + pn, 0.0f);
            partial += sqrtf(d);
        }
    }

    // ---- Block reduction, then one atomic add of the scaled partial ----
    lds_red[tid] = partial;
    __syncthreads();
    for (int s = 128; s > 0; s >>= 1) {
        if (tid < s) lds_red[tid] += lds_red[tid + s];
        __syncthreads();
    }
    if (tid == 0)
        atomicAdd(out, lds_red[0] * (0.5f / ((float)BATCH * (float)NPTS)));
}

extern "C" void kernel_launch(void* const* d_in, const int* in_sizes, int n_in,
                              void* d_out, int out_size, void* d_ws, size_t ws_size,
                              hipStream_t stream) {
    const float* pcd_pred = (const float*)d_in[0];
    const float* gt       = (const float*)d_in[1];
    float* out            = (float*)d_out;

    zero_out_kernel<<<1, 1, 0, stream>>>(out);

    dim3 grid(NPTS / ROWS_PER_BLOCK, BATCH, 2);  // (32, 4, 2) = 256 blocks
    chamfer_dir_kernel<<<grid, 256, 0, stream>>>(pcd_pred, gt, out);
}